// ProST_71889162600455
// MI455X (gfx1250) — compile-verified
//
#include <hip/hip_runtime.h>
#include <math.h>

typedef __attribute__((ext_vector_type(2))) float v2f;
typedef __attribute__((ext_vector_type(8))) float v8f;

#define HPIX   256
#define WPIX   256
#define NSTEP  96
#define DVOX   128
#define NPIX   (HPIX * WPIX)

// ws layout (floats):
//  [0..8]   R row-major
//  [12]     dmin
//  [13]     dstep = (dmax - dmin) / (NSTEP - 1)
//  [14..16] pbase  (voxel-space base point: 64*(M@[0,0,src,1]) + 64)
//  [32 + ray*4 + c]  pdelta per ray (64 * (R @ dirN)), c in {0,1,2}
#define WS_HDR_FLOATS 32

// ---------------------------------------------------------------------------
// Kernel 1: scalar setup — transform matrix, corner ray-distance range, bases.
// ---------------------------------------------------------------------------
__global__ void prost_setup(const float* __restrict__ pose,
                            const float* __restrict__ corner,
                            const float* __restrict__ param,
                            float* __restrict__ ws) {
  if (threadIdx.x != 0) return;
  float rx = pose[0], ry = pose[1], rz = pose[2];
  float tx = pose[3], ty = pose[4], tz = pose[5];
  float cxr = cosf(rx), sxr = sinf(rx);
  float cyr = cosf(ry), syr = sinf(ry);
  float czr = cosf(rz), szr = sinf(rz);
  // R = Rz @ Ry @ Rx
  float R00 = czr * cyr;
  float R01 = czr * syr * sxr - szr * cxr;
  float R02 = czr * syr * cxr + szr * sxr;
  float R10 = szr * cyr;
  float R11 = szr * syr * sxr + czr * cxr;
  float R12 = szr * syr * cxr - czr * sxr;
  float R20 = -syr;
  float R21 = cyr * sxr;
  float R22 = cyr * cxr;
  // translation column of M = R @ t
  float ttx = R00 * tx + R01 * ty + R02 * tz;
  float tty = R10 * tx + R11 * ty + R12 * tz;
  float ttz = R20 * tx + R21 * ty + R22 * tz;

  float srcv = param[0];

  // ray-distance range over the 8 corners: inv_pt = R @ (corner - t)
  float dmin = 1e30f, dmax = -1e30f;
  for (int c = 0; c < 8; ++c) {
    float p0 = corner[c * 3 + 0] - ttx;
    float p1 = corner[c * 3 + 1] - tty;
    float p2 = corner[c * 3 + 2] - ttz;
    float ix = R00 * p0 + R01 * p1 + R02 * p2;
    float iy = R10 * p0 + R11 * p1 + R12 * p2;
    float iz = R20 * p0 + R21 * p1 + R22 * p2;
    iz = srcv - iz;
    float dist = sqrtf(ix * ix + iy * iy + iz * iz);
    dmin = fminf(dmin, dist);
    dmax = fmaxf(dmax, dist);
  }

  // world base point = M @ [0,0,src,1] = src*R[:,2] + t ; to voxel space
  float bx = srcv * R02 + ttx;
  float by = srcv * R12 + tty;
  float bz = srcv * R22 + ttz;

  ws[0] = R00; ws[1] = R01; ws[2] = R02;
  ws[3] = R10; ws[4] = R11; ws[5] = R12;
  ws[6] = R20; ws[7] = R21; ws[8] = R22;
  ws[12] = dmin;
  ws[13] = (dmax - dmin) * (1.0f / (float)(NSTEP - 1));
  ws[14] = 64.0f * bx + 64.0f;
  ws[15] = 64.0f * by + 64.0f;
  ws[16] = 64.0f * bz + 64.0f;
}

// ---------------------------------------------------------------------------
// Kernel 2: per-ray direction transform delta = R @ dirN via
// V_WMMA_F32_16X16X4_F32. One wave handles 32 rays as two 16x4 @ 4x16 tiles.
// A layout (16x4 f32): lanes 0-15 -> rows M=0..15, VGPR0=K0, VGPR1=K1;
//                      lanes 16-31 -> same rows, VGPR0=K2, VGPR1=K3.
// B layout (4x16 f32): VGPR v, lane L: N = L%16, K = v + 2*(L>=16).
// C/D layout (16x16 f32): lane L: N = L%16, VGPR v: M = v + 8*(L>=16).
// ---------------------------------------------------------------------------
__global__ __launch_bounds__(256) void prost_dirs_wmma(
    const float* __restrict__ param,
    const float* __restrict__ ws,
    float* __restrict__ wsDelta) {
  const float det = param[1];
  const float pix = param[2];
  const int lane = threadIdx.x & 31;
  const int wave = (blockIdx.x * blockDim.x + threadIdx.x) >> 5;
  const bool hi = lane >= 16;
  const int n = lane & 15;

  // B: columns n<3 hold rows of R (B[k][n] = R[n][k]); rest zero.
  v2f b; b.x = 0.0f; b.y = 0.0f;
  if (n < 3) {
    b.x = ws[n * 3 + (hi ? 2 : 0)];   // K = 0 or 2
    b.y = hi ? 0.0f : ws[n * 3 + 1];  // K = 1 or 3 (K=3 column is zero pad)
  }

  for (int t = 0; t < 2; ++t) {
    const int tb = wave * 32 + t * 16;
    const int ray = tb + n;                  // this lane supplies A row M=n
    const int i = ray >> 8;                  // H index (ys)
    const int j = ray & (WPIX - 1);          // W index (xs)
    const float dxm = ((float)j - 127.5f) * pix;
    const float dym = ((float)i - 127.5f) * pix;
    const float inv = 1.0f / sqrtf(dxm * dxm + dym * dym + det * det);
    const float dnx = dxm * inv;
    const float dny = dym * inv;
    const float dnz = -det * inv;

    v2f a;
    a.x = hi ? dnz : dnx;   // K = 2 : 0
    a.y = hi ? 0.0f : dny;  // K = 3 : 1
    v8f c = {};
    c = __builtin_amdgcn_wmma_f32_16x16x4_f32(
        /*neg_a=*/false, a, /*neg_b=*/false, b,
        /*c_mod=*/(short)0, c, /*reuse_a=*/false, /*reuse_b=*/false);

    if (n < 3) {
#pragma unroll
      for (int v = 0; v < 8; ++v) {
        const int m = v + (hi ? 8 : 0);
        wsDelta[(tb + m) * 4 + n] = 64.0f * c[v];  // voxel-space delta
      }
    }
  }
}

// ---------------------------------------------------------------------------
// Ray-march core: 96 steps of trilinear sampling into the 128^3 volume.
// Clip-then-weight semantics match the reference exactly; OOB -> 0; NaN/Inf
// coordinates clamp to valid indices (memory-safe) and contribute 0.
// ---------------------------------------------------------------------------
__device__ __forceinline__ float prost_ray_march(
    const float* __restrict__ vol,
    float pbx, float pby, float pbz,
    float dx, float dy, float dz,
    float dmin, float dstep) {
  float acc = 0.0f;
#pragma unroll 4
  for (int s = 0; s < NSTEP; ++s) {
    const float d = fmaf((float)s, dstep, dmin);
    const float px = fmaf(d, dx, pbx);
    const float py = fmaf(d, dy, pby);
    const float pz = fmaf(d, dz, pbz);
    const bool inb = (px >= 0.0f) & (px <= 128.0f) &
                     (py >= 0.0f) & (py <= 128.0f) &
                     (pz >= 0.0f) & (pz <= 128.0f);
    const float fx = floorf(px), fy = floorf(py), fz = floorf(pz);
    const float x0 = fminf(fmaxf(fx, 0.0f), 127.0f);
    const float x1 = fminf(fmaxf(fx + 1.0f, 0.0f), 127.0f);
    const float y0 = fminf(fmaxf(fy, 0.0f), 127.0f);
    const float y1 = fminf(fmaxf(fy + 1.0f, 0.0f), 127.0f);
    const float z0 = fminf(fmaxf(fz, 0.0f), 127.0f);
    const float z1 = fminf(fmaxf(fz + 1.0f, 0.0f), 127.0f);
    const int ix0 = (int)x0, ix1 = (int)x1;
    const int iy0 = (int)y0, iy1 = (int)y1;
    const int iz0 = (int)z0, iz1 = (int)z1;
    const int b00 = (iz0 * DVOX + iy0) * DVOX;
    const int b01 = (iz0 * DVOX + iy1) * DVOX;
    const int b10 = (iz1 * DVOX + iy0) * DVOX;
    const int b11 = (iz1 * DVOX + iy1) * DVOX;
    const float Ia = vol[b00 + ix0], Ib = vol[b00 + ix1];
    const float Ic = vol[b01 + ix0], Id = vol[b01 + ix1];
    const float Ie = vol[b10 + ix0], If = vol[b10 + ix1];
    const float Ig = vol[b11 + ix0], Ih = vol[b11 + ix1];
    const float wx1 = x1 - px, wx0 = px - x0;
    const float wy1 = y1 - py, wy0 = py - y0;
    const float wz1 = z1 - pz, wz0 = pz - z0;
    const float v =
        wz1 * (wy1 * (wx1 * Ia + wx0 * Ib) + wy0 * (wx1 * Ic + wx0 * Id)) +
        wz0 * (wy1 * (wx1 * Ie + wx0 * If) + wy0 * (wx1 * Ig + wx0 * Ih));
    acc += inb ? v : 0.0f;
  }
  return acc;
}

// ---------------------------------------------------------------------------
// Kernel 3: one thread per output pixel, deltas precomputed by WMMA pass.
// ---------------------------------------------------------------------------
__global__ __launch_bounds__(256) void prost_gather(
    const float* __restrict__ vol,
    const float* __restrict__ wsHdr,
    const float* __restrict__ wsDelta,
    float* __restrict__ out) {
  const int pixId = blockIdx.x * blockDim.x + threadIdx.x;
  const float dminv = wsHdr[12], dstep = wsHdr[13];
  const float pbx = wsHdr[14], pby = wsHdr[15], pbz = wsHdr[16];
  const float dx = wsDelta[pixId * 4 + 0];
  const float dy = wsDelta[pixId * 4 + 1];
  const float dz = wsDelta[pixId * 4 + 2];
  out[pixId] = prost_ray_march(vol, pbx, pby, pbz, dx, dy, dz, dminv, dstep);
}

// ---------------------------------------------------------------------------
// Kernel 3b (fallback if ws too small for the delta table): delta inline.
// ---------------------------------------------------------------------------
__global__ __launch_bounds__(256) void prost_gather_fused(
    const float* __restrict__ vol,
    const float* __restrict__ param,
    const float* __restrict__ wsHdr,
    float* __restrict__ out) {
  const int pixId = blockIdx.x * blockDim.x + threadIdx.x;
  const float det = param[1];
  const float pixsp = param[2];
  const int i = pixId >> 8;
  const int j = pixId & (WPIX - 1);
  const float dxm = ((float)j - 127.5f) * pixsp;
  const float dym = ((float)i - 127.5f) * pixsp;
  const float inv = 1.0f / sqrtf(dxm * dxm + dym * dym + det * det);
  const float dnx = dxm * inv, dny = dym * inv, dnz = -det * inv;
  const float dx = 64.0f * (wsHdr[0] * dnx + wsHdr[1] * dny + wsHdr[2] * dnz);
  const float dy = 64.0f * (wsHdr[3] * dnx + wsHdr[4] * dny + wsHdr[5] * dnz);
  const float dz = 64.0f * (wsHdr[6] * dnx + wsHdr[7] * dny + wsHdr[8] * dnz);
  const float dminv = wsHdr[12], dstep = wsHdr[13];
  const float pbx = wsHdr[14], pby = wsHdr[15], pbz = wsHdr[16];
  out[pixId] = prost_ray_march(vol, pbx, pby, pbz, dx, dy, dz, dminv, dstep);
}

// ---------------------------------------------------------------------------
extern "C" void kernel_launch(void* const* d_in, const int* in_sizes, int n_in,
                              void* d_out, int out_size, void* d_ws,
                              size_t ws_size, hipStream_t stream) {
  (void)in_sizes; (void)n_in; (void)out_size;
  const float* ct     = (const float*)d_in[0];  // (1,1,128,128,128)
  // d_in[1] = fixed image: unused by the reference computation
  const float* pose   = (const float*)d_in[2];  // (1,6)
  const float* corner = (const float*)d_in[3];  // (1,8,3)
  const float* param  = (const float*)d_in[4];  // [src, det, pix_spacing, _]
  float* out = (float*)d_out;                   // 256*256 floats
  float* ws  = (float*)d_ws;

  prost_setup<<<1, 32, 0, stream>>>(pose, corner, param, ws);

  const size_t need = (size_t)(WS_HDR_FLOATS + (size_t)NPIX * 4) * sizeof(float);
  if (ws_size >= need) {
    // WMMA pass: 65536 rays, 32 rays per wave (two 16x16x4 f32 tiles).
    prost_dirs_wmma<<<NPIX / 256, 256, 0, stream>>>(param, ws,
                                                    ws + WS_HDR_FLOATS);
    prost_gather<<<NPIX / 256, 256, 0, stream>>>(ct, ws, ws + WS_HDR_FLOATS,
                                                 out);
  } else {
    prost_gather_fused<<<NPIX / 256, 256, 0, stream>>>(ct, param, ws, out);
  }
}